// label_lookup_53137335386161
// MI455X (gfx1250) — compile-verified
//
#include <hip/hip_runtime.h>
#include <hip/hip_bf16.h>

#define NUM_LABELS 200000
#define HIDDEN     64
#define BATCH      2048

typedef float v2f __attribute__((ext_vector_type(2)));
typedef float v8f __attribute__((ext_vector_type(8)));

// ---------------------------------------------------------------------------
// Kernel 1: gather embedding rows (writes both workspace copy and the
// return_embed output), and zero the loss-reduction scratch.
// ---------------------------------------------------------------------------
__global__ void gather_init_kernel(const int* __restrict__ label,
                                   const float* __restrict__ table,
                                   float* __restrict__ ws_embed,
                                   float* __restrict__ out_embed,
                                   float* __restrict__ Z,
                                   float* __restrict__ sval) {
    int gid = blockIdx.x * 256 + threadIdx.x;
    if (gid < BATCH) { Z[gid] = 0.0f; sval[gid] = 0.0f; }
    if (gid < BATCH * HIDDEN) {
        int row = gid >> 6;           // / HIDDEN
        int col = gid & 63;           // % HIDDEN
        float v = table[(size_t)label[row] * HIDDEN + col];
        ws_embed[gid]  = v;
        out_embed[gid] = v;           // d_out region is only 4B aligned -> b32 stores
    }
}

// ---------------------------------------------------------------------------
// Kernel 2: fused  logits = sigmoid(embed @ W^T + b)  via V_WMMA_F32_16X16X4_F32
// plus per-row sum(exp(logit)) accumulation and label-column capture.
//
// Wave mapping (wave32): wave owns an M-tile of 16 rows; A (16x64) held in 16
// v2f fragments for the whole kernel. Per N-tile: batch-load all 16 B fragments
// (one s_clause of global_load_b64), then a 16-deep WMMA accumulation chain,
// then the sigmoid/softmax epilogue with non-temporal stores.
// Block = 8 waves sharing one N strip (B reuse through WGP$).
// Grid = (128/8 = 16 M-groups, 200000/(20*16) = 625 strips).
// ---------------------------------------------------------------------------
__global__ void __launch_bounds__(256)
gemm_sigmoid_kernel(const float* __restrict__ ws_embed,
                    const int*   __restrict__ label,
                    const float* __restrict__ W,
                    const float* __restrict__ bvec,
                    float*       __restrict__ logits,
                    float*       __restrict__ Z,
                    float*       __restrict__ sval) {
    const int lane   = threadIdx.x & 31;
    const int wid    = threadIdx.x >> 5;
    const int lanelo = lane & 15;
    const int hi     = lane >> 4;

    const int mtile = blockIdx.x * 8 + wid;   // 0..127
    const int mbase = mtile * 16;

    // --- load full A fragment set: 16x64 f32 tile, 16 ksteps of 16x16x4 ---
    // f32 A layout: lanes 0-15 -> M=lane, K = {4k, 4k+1}; lanes 16-31 -> K = {4k+2, 4k+3}
    v2f afrag[16];
    const float* aptr = ws_embed + (size_t)(mbase + lanelo) * HIDDEN + 2 * hi;
    #pragma unroll
    for (int k = 0; k < 16; ++k)
        afrag[k] = *reinterpret_cast<const v2f*>(aptr + 4 * k);

    // labels for the 16 rows this wave produces (C/D layout: VGPR v -> row v + 8*hi)
    int labv[8];
    #pragma unroll
    for (int v = 0; v < 8; ++v)
        labv[v] = label[mbase + v + 8 * hi];

    float zpart[8];
    #pragma unroll
    for (int v = 0; v < 8; ++v) zpart[v] = 0.0f;

    const int nstrip = blockIdx.y * 320;      // 20 tiles * 16 cols

    for (int t = 0; t < 20; ++t) {
        const int n0  = nstrip + t * 16;
        const int col = n0 + lanelo;

        // B = W^T: B[k][n] = W[n][k]; lane loads W[col] at K = {4k+2hi, 4k+2hi+1}.
        // Batch all 16 fragments -> single load clause, overlaps prior epilogue.
        const float* bptr = W + (size_t)col * HIDDEN + 2 * hi;
        v2f bfrag[16];
        #pragma unroll
        for (int k = 0; k < 16; ++k)
            bfrag[k] = *reinterpret_cast<const v2f*>(bptr + 4 * k);

        const float bias = bvec[col];

        v8f c;
        #pragma unroll
        for (int i = 0; i < 8; ++i) c[i] = 0.0f;

        #pragma unroll
        for (int k = 0; k < 16; ++k) {
            c = __builtin_amdgcn_wmma_f32_16x16x4_f32(
                    /*neg_a=*/false, afrag[k],
                    /*neg_b=*/false, bfrag[k],
                    /*c_mod=*/(short)0, c,
                    /*reuse_a=*/false, /*reuse_b=*/false);
        }

        // epilogue: bias + sigmoid (hw rcp), NT store, sum-exp, label capture
        #pragma unroll
        for (int v = 0; v < 8; ++v) {
            const int row = mbase + v + 8 * hi;
            float x = c[v] + bias;
            float s = __builtin_amdgcn_rcpf(1.0f + __expf(-x));   // v_rcp_f32
            __builtin_nontemporal_store(s, logits + (size_t)row * NUM_LABELS + col);
            zpart[v] += __expf(s);            // s in (0,1): no max-shift needed
            if (labv[v] == col) sval[row] = s;
        }
    }

    // reduce zpart across the 16 lanes holding one row, then one atomic per row
    #pragma unroll
    for (int v = 0; v < 8; ++v) {
        float z = zpart[v];
        #pragma unroll
        for (int m = 8; m >= 1; m >>= 1)
            z += __shfl_xor(z, m, 16);
        if (lanelo == 0)
            atomicAdd(&Z[mbase + v + 8 * hi], z);
    }
}

// ---------------------------------------------------------------------------
// Kernel 3: loss = -mean( sval[i] - log(Z[i]) )
// ---------------------------------------------------------------------------
__global__ void loss_final_kernel(const float* __restrict__ Z,
                                  const float* __restrict__ sval,
                                  float* __restrict__ out_loss) {
    __shared__ float red[256];
    float acc = 0.0f;
    for (int i = threadIdx.x; i < BATCH; i += 256)
        acc += sval[i] - __logf(Z[i]);
    red[threadIdx.x] = acc;
    __syncthreads();
    for (int s = 128; s > 0; s >>= 1) {
        if (threadIdx.x < s) red[threadIdx.x] += red[threadIdx.x + s];
        __syncthreads();
    }
    if (threadIdx.x == 0)
        *out_loss = -red[0] / (float)BATCH;
}

// ---------------------------------------------------------------------------
extern "C" void kernel_launch(void* const* d_in, const int* in_sizes, int n_in,
                              void* d_out, int out_size, void* d_ws, size_t ws_size,
                              hipStream_t stream) {
    const int*   label = (const int*)  d_in[0];   // [2048] int32
    const float* table = (const float*)d_in[1];   // [200000,64]
    const float* W     = (const float*)d_in[2];   // [200000,64]
    const float* bvec  = (const float*)d_in[3];   // [200000]

    float* out = (float*)d_out;
    float* logits    = out;                                     // [2048*200000]
    float* out_loss  = out + (size_t)BATCH * NUM_LABELS;        // [1]
    float* out_embed = out_loss + 1;                            // [2048*64]

    // workspace layout
    float* ws_embed = (float*)d_ws;                 // 131072 floats
    float* Z        = ws_embed + BATCH * HIDDEN;    // 2048 floats
    float* sval     = Z + BATCH;                    // 2048 floats

    // 1) gather + init scratch
    gather_init_kernel<<<(BATCH * HIDDEN + 255) / 256, 256, 0, stream>>>(
        label, table, ws_embed, out_embed, Z, sval);

    // 2) fused WMMA GEMM + sigmoid + softmax statistics
    dim3 grid(16, 625);   // 16*8 = 128 M-tiles ; 625*320 = 200000 cols
    gemm_sigmoid_kernel<<<grid, 256, 0, stream>>>(
        ws_embed, label, W, bvec, logits, Z, sval);

    // 3) scalar loss
    loss_final_kernel<<<1, 256, 0, stream>>>(Z, sval, out_loss);
}